// HyenaBlock_2929167696304
// MI455X (gfx1250) — compile-verified
//
#include <hip/hip_runtime.h>
#include <hip/hip_bf16.h>
#include <math.h>
#include <stdint.h>

#define Bn 4
#define Sn 8192
#define Dn 128
#define Nn2 2
#define Kc 3
#define Gn 3
#define BS (Bn*Sn)          // 32768
#define EPS_L1 1e-8f

typedef float v2f __attribute__((ext_vector_type(2)));
typedef float v8f __attribute__((ext_vector_type(8)));
typedef int   v4i_ __attribute__((vector_size(16)));
typedef __attribute__((address_space(1))) v4i_* as1_v4i;
typedef __attribute__((address_space(3))) v4i_* as3_v4i;

#if defined(__has_builtin)
#  if __has_builtin(__builtin_amdgcn_global_load_async_to_lds_b128)
#    define HAVE_ASYNC_BUILTIN 1
#  endif
#endif

// 16-byte global -> LDS async copy (ASYNCcnt-tracked, no VGPR round trip)
__device__ __forceinline__ void async_ld16(float* lds, const float* g) {
#ifdef HAVE_ASYNC_BUILTIN
    __builtin_amdgcn_global_load_async_to_lds_b128(
        (as1_v4i)g, (as3_v4i)lds, 0, 0);
#else
    unsigned           lo = (unsigned)(unsigned long long)(uintptr_t)lds;
    unsigned long long ga = (unsigned long long)(uintptr_t)g;
    asm volatile("global_load_async_to_lds_b128 %0, %1, off"
                 :: "v"(lo), "v"(ga) : "memory");
#endif
}

__device__ __forceinline__ void wait_async0() {
    asm volatile("s_wait_asynccnt 0x0" ::: "memory");
}

// ---------------------------------------------------------------------------
// Generic f32 WMMA GEMM: C[M,N] = (A[M,K] @ B[K,N] + bias) * scale
// Block = 256 threads = 8 waves; waves tiled 2 (M) x 4 (N); each wave 16x16.
// Uses V_WMMA_F32_16X16X4_F32, K stepped by 4.
// A fragment layout (ISA 7.12.2, 32-bit A 16x4): lanes 0-15: K=0,1 ; 16-31: K=2,3
// B fragment: lanes 0-15: rows K=0,1 ; lanes 16-31: rows K=2,3 (cols = lane%16)
// C/D: vgpr r -> M=r (lanes 0-15), M=8+r (lanes 16-31); N = lane%16
// ---------------------------------------------------------------------------
__global__ void gemm_f32_wmma(const float* __restrict__ A,
                              const float* __restrict__ Bm,
                              const float* __restrict__ bias,
                              const float* __restrict__ scale,
                              float* __restrict__ C,
                              int M, int Nc, int Kk) {
    const int lane = threadIdx.x & 31;
    const int w    = threadIdx.x >> 5;
    const int mt   = blockIdx.x * 32 + (w & 1) * 16;
    const int nt   = blockIdx.y * 64 + (w >> 1) * 16;
    const int lm   = lane & 15;
    const int kh   = (lane >> 4) << 1;      // 0 or 2

    v8f acc = {0.f,0.f,0.f,0.f,0.f,0.f,0.f,0.f};

    const float* arow = A  + (size_t)(mt + lm) * Kk + kh;
    const float* bcol = Bm + (size_t)kh * Nc + nt + lm;

#pragma unroll 8
    for (int k = 0; k < Kk; k += 4) {
        v2f a = *(const v2f*)(arow + k);                 // A[m, k+kh], A[m, k+kh+1]
        v2f b;
        b.x = bcol[(size_t)k * Nc];                      // B[k+kh,   n]
        b.y = bcol[(size_t)k * Nc + Nc];                 // B[k+kh+1, n]
        acc = __builtin_amdgcn_wmma_f32_16x16x4_f32(
                  false, a, false, b, (short)0, acc, false, false);
    }

    const int  col = nt + lm;
    const float bb = bias  ? bias[col]  : 0.0f;
    const float sc = scale ? scale[col] : 1.0f;
    const int  mr  = mt + ((lane >> 4) << 3);
#pragma unroll
    for (int r = 0; r < 8; ++r) {
        C[(size_t)(mr + r) * Nc + col] = (acc[r] + bb) * sc;
    }
}

// ---------------------------------------------------------------------------
// Depthwise conv (K=3, same pad) + the reference's flat reshape
// (B,S,384) -> (B,D,G,S). Output layout zg[b][g][d][s] = (B,G,D,S).
// f = d*G*S + g*S + s ; s2 = f/384 ; c = f%384 ; value = conv(x[b,:,c]) at s2
// ---------------------------------------------------------------------------
__global__ void conv_gather(const float* __restrict__ X,
                            const float* __restrict__ cw,
                            const float* __restrict__ cb,
                            float* __restrict__ zg) {
    const int tid = blockIdx.x * blockDim.x + threadIdx.x;   // B*G*D*S threads
    const int s    = tid & (Sn - 1);
    int rem        = tid >> 13;          // /S
    const int d    = rem & (Dn - 1);
    rem          >>= 7;                  // /D
    const int g    = rem % Gn;
    const int b    = rem / Gn;

    const int f  = d * (Gn * Sn) + g * Sn + s;
    const int s2 = f / (Gn * Dn);
    const int c  = f - s2 * (Gn * Dn);

    const float w0 = cw[c * 3 + 0], w1 = cw[c * 3 + 1], w2 = cw[c * 3 + 2];
    const float* xb = X + (size_t)b * Sn * (Gn * Dn);
    const float xm = (s2 > 0)      ? xb[(size_t)(s2 - 1) * (Gn * Dn) + c] : 0.f;
    const float x0 =                 xb[(size_t)s2       * (Gn * Dn) + c];
    const float xp = (s2 < Sn - 1) ? xb[(size_t)(s2 + 1) * (Gn * Dn) + c] : 0.f;

    zg[tid] = w0 * xm + w1 * x0 + w2 * xp + cb[c];
}

// ---------------------------------------------------------------------------
// RoPE: t[r, 2j] / t[r, 2j+1] from embeddings + positions. 1 thread / pair.
// ---------------------------------------------------------------------------
__global__ void rope_kernel(const float* __restrict__ E,
                            const int* __restrict__ pos,
                            float* __restrict__ T) {
    const int tid = blockIdx.x * blockDim.x + threadIdx.x;   // BS*64 threads
    const int j = tid & 63;
    const int r = tid >> 6;
    const float theta = __expf(-(float)j * (2.0f / 128.0f) * 9.210340371976184f); // ln(1e4)
    const float ang = (float)pos[r] * theta;
    float sv, cv;
    sincosf(ang, &sv, &cv);
    const float xe = E[(size_t)r * Dn + 2 * j];
    const float xo = E[(size_t)r * Dn + 2 * j + 1];
    T[(size_t)r * Dn + 2 * j]     = xe * cv - xo * sv;
    T[(size_t)r * Dn + 2 * j + 1] = xe * sv + xo * cv;
}

// ---------------------------------------------------------------------------
// LayerNorm(D=128) + exact GELU. One wave32 per row, 4 f32 per lane.
// ---------------------------------------------------------------------------
__global__ void ln_gelu_kernel(const float* __restrict__ Hraw,
                               const float* __restrict__ g,
                               const float* __restrict__ bv,
                               float* __restrict__ Hmid) {
    const int lane = threadIdx.x & 31;
    const int w    = threadIdx.x >> 5;
    const int r    = blockIdx.x * 8 + w;
    const float4 x = *(const float4*)(Hraw + (size_t)r * Dn + lane * 4);
    float s1 = x.x + x.y + x.z + x.w;
    float s2 = x.x * x.x + x.y * x.y + x.z * x.z + x.w * x.w;
#pragma unroll
    for (int m = 1; m < 32; m <<= 1) {
        s1 += __shfl_xor(s1, m, 32);
        s2 += __shfl_xor(s2, m, 32);
    }
    const float mu = s1 * (1.0f / 128.0f);
    const float var = s2 * (1.0f / 128.0f) - mu * mu;
    const float rs = rsqrtf(var + 1e-5f);
    float4 o;
    const float* gp = g + lane * 4;
    const float* bp = bv + lane * 4;
    float xs[4] = {x.x, x.y, x.z, x.w};
    float os[4];
#pragma unroll
    for (int q = 0; q < 4; ++q) {
        float y = (xs[q] - mu) * rs * gp[q] + bp[q];
        os[q] = 0.5f * y * (1.0f + erff(y * 0.70710678118654752f));
    }
    o.x = os[0]; o.y = os[1]; o.z = os[2]; o.w = os[3];
    *(float4*)(Hmid + (size_t)r * Dn + lane * 4) = o;
}

// ---------------------------------------------------------------------------
// L1 normalize over D per half, with transpose to (B, N, D, S).
// One wave32 per row of h2 (256 cols); lane handles 8 cols within one half.
// ---------------------------------------------------------------------------
__global__ void l1norm_kernel(const float* __restrict__ H2,
                              float* __restrict__ HN) {
    const int lane = threadIdx.x & 31;
    const int w    = threadIdx.x >> 5;
    const int r    = blockIdx.x * 8 + w;
    const int b = r >> 13;
    const int s = r & (Sn - 1);
    const float* row = H2 + (size_t)r * (Nn2 * Dn) + lane * 8;
    float v[8], asum = 0.f;
#pragma unroll
    for (int q = 0; q < 8; ++q) { v[q] = row[q]; asum += fabsf(v[q]); }
#pragma unroll
    for (int m = 1; m < 16; m <<= 1) asum += __shfl_xor(asum, m, 32); // within half (16 lanes)
    const float inv = 1.0f / (asum + EPS_L1);
#pragma unroll
    for (int q = 0; q < 8; ++q) {
        const int c = lane * 8 + q;
        const int i = c >> 7;
        const int d = c & (Dn - 1);
        HN[(((size_t)(b * Nn2 + i)) * Dn + d) * Sn + s] = v[q] * inv;
    }
}

// ---------------------------------------------------------------------------
// One Hyena recurrence step: vout = z_g * ( (1/2S) causal_conv(v, h_i) + Bp_i*v )
// One block per (b,d) row. v,h staged into 64KB dynamic LDS via
// GLOBAL_LOAD_ASYNC_TO_LDS_B128 (ASYNCcnt). 8x8 register blocking:
// 64 FMAs per 23 LDS reads.
// ---------------------------------------------------------------------------
__global__ void conv_step(const float* __restrict__ vin, int vinG, int vinSel,
                          const float* __restrict__ hn, int hi,
                          const float* __restrict__ zg, int gsel,
                          const float* __restrict__ Bp,
                          float* __restrict__ vout) {
    extern __shared__ float smem[];
    float* lv = smem;
    float* lh = smem + Sn;

    const int b = blockIdx.x >> 7;          // /D
    const int d = blockIdx.x & (Dn - 1);
    const float* vrow = vin + (((size_t)b * vinG + vinSel) * Dn + d) * Sn;
    const float* hrow = hn  + (((size_t)b * Nn2 + hi)      * Dn + d) * Sn;
    const float* zrow = zg  + (((size_t)b * Gn  + gsel)    * Dn + d) * Sn;
    float* orow       = vout + ((size_t)b * Dn + d) * Sn;
    const float bp = Bp[hi * Dn + d];

    // async stage v and h rows into LDS (16B per lane per op)
#pragma unroll
    for (int i = threadIdx.x * 4; i < Sn; i += 256 * 4) {
        async_ld16(lv + i, vrow + i);
        async_ld16(lh + i, hrow + i);
    }
    // warm L2/WGP$ for the epilogue operand while we compute
    __builtin_prefetch(zrow + threadIdx.x * 32, 0, 1);
    wait_async0();
    __syncthreads();

    const float inv = 1.0f / (2.0f * Sn);

    for (int task = threadIdx.x; task < Sn / 8; task += 256) {
        const int t0 = task * 8;
        float acc[8] = {0.f,0.f,0.f,0.f,0.f,0.f,0.f,0.f};

        int tau0 = 0;
        for (; tau0 <= t0 - 8; tau0 += 8) {
            float v8[8], hw[15];
            const int hb = t0 - tau0 - 7;            // >= 1 in this loop
#pragma unroll
            for (int j = 0; j < 8; ++j)  v8[j] = lv[tau0 + j];
#pragma unroll
            for (int q = 0; q < 15; ++q) hw[q] = lh[hb + q];
#pragma unroll
            for (int j = 0; j < 8; ++j)
#pragma unroll
                for (int r = 0; r < 8; ++r)
                    acc[r] = fmaf(v8[j], hw[7 + r - j], acc[r]);
        }
        // triangular tail
        for (int tau = tau0; tau <= t0 + 7; ++tau) {
            const float vv = lv[tau];
#pragma unroll
            for (int r = 0; r < 8; ++r) {
                const int h = t0 + r - tau;
                if (h >= 0) acc[r] = fmaf(vv, lh[h], acc[r]);
            }
        }
#pragma unroll
        for (int r = 0; r < 8; ++r) {
            const int t = t0 + r;
            const float u = acc[r] * inv + bp * lv[t];
            orow[t] = zrow[t] * u;
        }
    }
}

// ---------------------------------------------------------------------------
// (B, D, S) -> (B, S, D) tiled transpose.
// ---------------------------------------------------------------------------
__global__ void transpose_ds(const float* __restrict__ V, float* __restrict__ VT) {
    __shared__ float tile[32][33];
    const int b  = blockIdx.z;
    const int s0 = blockIdx.x * 32;
    const int d0 = blockIdx.y * 32;
    const int tx = threadIdx.x, ty = threadIdx.y;
#pragma unroll
    for (int i = 0; i < 32; i += 8)
        tile[ty + i][tx] = V[((size_t)b * Dn + d0 + ty + i) * Sn + s0 + tx];
    __syncthreads();
#pragma unroll
    for (int i = 0; i < 32; i += 8)
        VT[((size_t)b * Sn + s0 + ty + i) * Dn + d0 + tx] = tile[tx][ty + i];
}

// ---------------------------------------------------------------------------
extern "C" void kernel_launch(void* const* d_in, const int* in_sizes, int n_in,
                              void* d_out, int out_size, void* d_ws, size_t ws_size,
                              hipStream_t stream) {
    const float* E     = (const float*)d_in[0];
    const float* Wproj = (const float*)d_in[1];
    const float* convw = (const float*)d_in[2];
    const float* convb = (const float*)d_in[3];
    const float* W1    = (const float*)d_in[4];
    const float* b1    = (const float*)d_in[5];
    const float* lng   = (const float*)d_in[6];
    const float* lnb   = (const float*)d_in[7];
    const float* W2    = (const float*)d_in[8];
    const float* b2    = (const float*)d_in[9];
    const float* ffs   = (const float*)d_in[10];
    const float* outp  = (const float*)d_in[11];
    const float* Bpar  = (const float*)d_in[12];
    const int*   pos   = (const int*)d_in[13];
    float* out = (float*)d_out;
    float* ws  = (float*)d_ws;

    // workspace layout (floats); regions reused once dead
    float* X  = ws;                         // 12,582,912  x = E@Wproj ; later h2 ; later vT
    float* Z  = ws + 12582912ull;           // 12,582,912  zg (B,G,D,S)
    float* T  = ws + 25165824ull;           //  4,194,304  t ; later hmid ; later v2
    float* H  = ws + 29360128ull;           //  4,194,304  hraw ; later v1
    float* HN = ws + 33554432ull;           //  8,388,608  h_hat normalized (B,N,D,S)
    float* H2 = X;
    float* V1 = H;
    float* V2 = T;
    float* VT = X;

    const dim3 blk(256);

    // 1) x = E @ W_proj                       (32768 x 384, K=128)
    gemm_f32_wmma<<<dim3(BS/32, 384/64), blk, 0, stream>>>(E, Wproj, nullptr, nullptr, X, BS, 384, Dn);
    // 2) depthwise conv + flat reshape -> zg
    conv_gather<<<(Bn*Gn*Dn*Sn)/256, blk, 0, stream>>>(X, convw, convb, Z);
    // 3) t = rope(E, pos)
    rope_kernel<<<(BS*64)/256, blk, 0, stream>>>(E, pos, T);
    // 4) hraw = t @ W1 + b1
    gemm_f32_wmma<<<dim3(BS/32, 128/64), blk, 0, stream>>>(T, W1, b1, nullptr, H, BS, Dn, Dn);
    // 5) hmid = gelu(LN(hraw))
    ln_gelu_kernel<<<BS/8, blk, 0, stream>>>(H, lng, lnb, T);
    // 6) h2 = (hmid @ W2 + b2) * ff_scale
    gemm_f32_wmma<<<dim3(BS/32, 256/64), blk, 0, stream>>>(T, W2, b2, ffs, H2, BS, Nn2*Dn, Dn);
    // 7) h_hat: transpose + L1 normalize over D
    l1norm_kernel<<<BS/8, blk, 0, stream>>>(H2, HN);
    // 8) recurrence: v = z2 ; i=0: v1 = z0*(conv(v,h0)+B0*v) ; i=1: v2 = z1*(conv(v1,h1)+B1*v1)
    conv_step<<<Bn*Dn, blk, 2*Sn*sizeof(float), stream>>>(Z, Gn, Gn-1, HN, 0, Z, 0, Bpar, V1);
    conv_step<<<Bn*Dn, blk, 2*Sn*sizeof(float), stream>>>(V1, 1, 0,    HN, 1, Z, 1, Bpar, V2);
    // 9) vT (B,S,D)
    transpose_ds<<<dim3(Sn/32, Dn/32, Bn), dim3(32, 8), 0, stream>>>(V2, VT);
    // 10) out = vT @ out_proj
    gemm_f32_wmma<<<dim3(BS/32, 128/64), blk, 0, stream>>>(VT, outp, nullptr, nullptr, out, BS, Dn, Dn);
}